// MeshSDFLoss_81260781240476
// MI455X (gfx1250) — compile-verified
//
#include <hip/hip_runtime.h>
#include <hip/hip_bf16.h>

// MeshSDF brute-force closest-point loss for gfx1250 (MI455X, wave32).
//
// Per-pair math reduces to three dots (s1=ab.p, s2=ac.p, s3=a.p) computed by
// V_WMMA_F32_16X16X4_F32 (16 points x K=4 x 16 triangle columns), plus
// per-triangle constants. F-dimension split 4-way for occupancy (2048 waves).

typedef __attribute__((ext_vector_type(2))) float v2f;
typedef __attribute__((ext_vector_type(4))) float v4f;
typedef __attribute__((ext_vector_type(8))) float v8f;

#define EPS 1e-12f
#define FSPLIT 4

// ---------------------------------------------------------------------------
// Prep: per face, pack WMMA B-operand blocks (lane-exact layout) + constants.
// B (4x16 f32) layout (row-striped like C/D, split across lane halves):
//   VGPR0: lanes 0-15 = row K=0, lanes 16-31 = row K=2
//   VGPR1: lanes 0-15 = row K=1, lanes 16-31 = row K=3
// Per tri-tile t: 32-lane x 2-float block; lane n (n<16) = (v.x,v.y) of
// column n, lane n+16 = (v.z, 0).
// Constants per tri (8 floats): kaba,kaca,kbb,kcc,kbc,kaa,rkbb,rkcc.
// ---------------------------------------------------------------------------
__global__ void meshsdf_prep(const float* __restrict__ verts,
                             const int* __restrict__ faces,
                             float* __restrict__ wsAB,
                             float* __restrict__ wsAC,
                             float* __restrict__ wsAV,
                             float* __restrict__ wsK,
                             int F) {
    int j = blockIdx.x * blockDim.x + threadIdx.x;
    if (j >= F) return;
    int ia = faces[3 * j + 0], ib = faces[3 * j + 1], ic = faces[3 * j + 2];
    float ax = verts[3 * ia + 0], ay = verts[3 * ia + 1], az = verts[3 * ia + 2];
    float bx = verts[3 * ib + 0], by = verts[3 * ib + 1], bz = verts[3 * ib + 2];
    float cx = verts[3 * ic + 0], cy = verts[3 * ic + 1], cz = verts[3 * ic + 2];
    float abx = bx - ax, aby = by - ay, abz = bz - az;
    float acx = cx - ax, acy = cy - ay, acz = cz - az;

    float kaba = abx * ax + aby * ay + abz * az;
    float kaca = acx * ax + acy * ay + acz * az;
    float kbb  = abx * abx + aby * aby + abz * abz;
    float kcc  = acx * acx + acy * acy + acz * acz;
    float kbc  = abx * acx + aby * acy + abz * acz;
    float kaa  = ax * ax + ay * ay + az * az;

    int t = j >> 4, n = j & 15;
    float* pab = wsAB + t * 64;
    float* pac = wsAC + t * 64;
    float* pav = wsAV + t * 64;
    pab[2 * n + 0] = abx; pab[2 * n + 1] = aby;
    pab[2 * (n + 16) + 0] = abz; pab[2 * (n + 16) + 1] = 0.0f;
    pac[2 * n + 0] = acx; pac[2 * n + 1] = acy;
    pac[2 * (n + 16) + 0] = acz; pac[2 * (n + 16) + 1] = 0.0f;
    pav[2 * n + 0] = ax; pav[2 * n + 1] = ay;
    pav[2 * (n + 16) + 0] = az; pav[2 * (n + 16) + 1] = 0.0f;

    float* pk = wsK + 8 * j;
    pk[0] = kaba; pk[1] = kaca; pk[2] = kbb; pk[3] = kcc;
    pk[4] = kbc;  pk[5] = kaa;
    pk[6] = 1.0f / fmaxf(kbb, EPS);   // exact; used as t_ab = d1 * rkbb
    pk[7] = 1.0f / fmaxf(kcc, EPS);
}

// ---------------------------------------------------------------------------
// Main: wave w -> (point strip pt = w/FSPLIT, segment s = w%FSPLIT).
// Each wave owns 16 points and scans ntile/FSPLIT triangle tiles.
// Per tile: 3x V_WMMA_F32_16X16X4_F32 (S1,S2,S3) with register double-buffered
// B operands, then Ericson region select per pair, running min/argmin per
// lane, cross-lane reduce, candidate written to workspace.
// ---------------------------------------------------------------------------
__global__ __launch_bounds__(128) void meshsdf_main(
    const float* __restrict__ points,
    const float* __restrict__ wsAB,
    const float* __restrict__ wsAC,
    const float* __restrict__ wsAV,
    const float* __restrict__ wsK,
    float* __restrict__ candD,          // [FSPLIT][P]
    int* __restrict__ candI,            // [FSPLIT][P]
    int P, int F) {
    const int lane = threadIdx.x & 31;
    const int wave = blockIdx.x * (blockDim.x >> 5) + (threadIdx.x >> 5);
    const int pt   = wave / FSPLIT;     // point strip
    const int seg  = wave % FSPLIT;     // triangle segment
    const int pbase = pt * 16;
    if (pbase >= P) return;             // wave-uniform
    const bool hi = lane >= 16;
    const int row = lane & 15;

    // A operand (16x4 f32): lanes 0-15 -> (x,y) K=0,1 ; lanes 16-31 -> (z,0).
    const float* pp = points + 3 * (pbase + row);
    v2f A;
    A.x = hi ? pp[2] : pp[0];
    A.y = hi ? 0.0f : pp[1];

    // |p|^2 per point, computed by lanes 0..15, broadcast into pn2[r].
    int pidx = pbase + lane; if (pidx >= P) pidx = P - 1;
    const float* pq = points + 3 * pidx;
    float qn = pq[0] * pq[0] + pq[1] * pq[1] + pq[2] * pq[2];
    const int src0 = hi ? 8 : 0;
    float pn2[8];
#pragma unroll
    for (int r = 0; r < 8; ++r) pn2[r] = __shfl(qn, src0 + r, 32);

    float rmin[8];
    int   ridx[8];
#pragma unroll
    for (int r = 0; r < 8; ++r) { rmin[r] = 3.4e38f; ridx[r] = 0; }

    const int ntile  = F >> 4;
    const int tstart = (ntile * seg) / FSPLIT;
    const int tend   = (ntile * (seg + 1)) / FSPLIT;

    // Software-pipelined B operands (register double buffer).
    v2f Bab = *(const v2f*)(wsAB + tstart * 64 + 2 * lane);
    v2f Bac = *(const v2f*)(wsAC + tstart * 64 + 2 * lane);
    v2f Bav = *(const v2f*)(wsAV + tstart * 64 + 2 * lane);

    for (int t = tstart; t < tend; ++t) {
        // Issue next-tile loads before the heavy VALU epilogue.
        v2f nAB = Bab, nAC = Bac, nAV = Bav;
        if (t + 1 < tend) {
            nAB = *(const v2f*)(wsAB + (t + 1) * 64 + 2 * lane);
            nAC = *(const v2f*)(wsAC + (t + 1) * 64 + 2 * lane);
            nAV = *(const v2f*)(wsAV + (t + 1) * 64 + 2 * lane);
        }
        const int j = t * 16 + row;          // this lane's triangle column
        v4f k0 = *(const v4f*)(wsK + 8 * j);
        v4f k1 = *(const v4f*)(wsK + 8 * j + 4);

        v8f z = {0.f, 0.f, 0.f, 0.f, 0.f, 0.f, 0.f, 0.f};
        // D = A x B + 0 : 16 points x 16 triangle columns of dot products.
        v8f S1 = __builtin_amdgcn_wmma_f32_16x16x4_f32(false, A, false, Bab,
                                                       (short)0, z, false, false);
        v8f S2 = __builtin_amdgcn_wmma_f32_16x16x4_f32(false, A, false, Bac,
                                                       (short)0, z, false, false);
        v8f S3 = __builtin_amdgcn_wmma_f32_16x16x4_f32(false, A, false, Bav,
                                                       (short)0, z, false, false);

        const float kaba = k0.x, kaca = k0.y, kbb = k0.z, kcc = k0.w;
        const float kbc = k1.x, kaa = k1.y, rkbb = k1.z, rkcc = k1.w;

#pragma unroll
        for (int r = 0; r < 8; ++r) {
            float s1 = S1[r], s2 = S2[r], s3 = S3[r];
            float d1 = s1 - kaba, d2 = s2 - kaca;
            float d3 = d1 - kbb, d4 = d2 - kbc, d5 = d1 - kbc, d6 = d2 - kcc;
            float vc = d1 * d4 - d3 * d2;
            float vb = d5 * d2 - d1 * d6;
            float va = d3 * d6 - d5 * d4;
            float t_ab = d1 * rkbb;                 // d1-d3 == kbb
            float t_ac = d2 * rkcc;                 // d2-d6 == kcc
            float e43 = d4 - d3, e56 = d5 - d6;
            float t_bc = e43 * __builtin_amdgcn_rcpf(fmaxf(e43 + e56, EPS));
            float rden = __builtin_amdgcn_rcpf(fmaxf(va + vb + vc, EPS));
            float bv = vb * rden, bw = vc * rden;
            // Region selects in reverse priority (later overrides earlier).
            bool cBC = (va <= 0.f) & (e43 >= 0.f) & (e56 >= 0.f);
            bv = cBC ? 1.f - t_bc : bv; bw = cBC ? t_bc : bw;
            bool cAC = (vb <= 0.f) & (d2 >= 0.f) & (d6 <= 0.f);
            bv = cAC ? 0.f : bv;        bw = cAC ? t_ac : bw;
            bool cC = (d6 >= 0.f) & (d5 <= d6);
            bv = cC ? 0.f : bv;         bw = cC ? 1.f : bw;
            bool cAB = (vc <= 0.f) & (d1 >= 0.f) & (d3 <= 0.f);
            bv = cAB ? t_ab : bv;       bw = cAB ? 0.f : bw;
            bool cB = (d3 >= 0.f) & (d4 <= d3);
            bv = cB ? 1.f : bv;         bw = cB ? 0.f : bw;
            bool cA = (d1 <= 0.f) & (d2 <= 0.f);
            bv = cA ? 0.f : bv;         bw = cA ? 0.f : bw;
            // dist^2 = |p|^2 - 2 p.cp + |cp|^2, all from precomputed dots.
            float dd = pn2[r] - 2.f * (s3 + bv * s1 + bw * s2)
                     + (kaa + bv * bv * kbb + bw * bw * kcc
                        + 2.f * (bv * kaba + bw * kaca + bv * bw * kbc));
            dd = fmaxf(dd, 0.f);
            if (dd < rmin[r]) { rmin[r] = dd; ridx[r] = j; }
        }
        Bab = nAB; Bac = nAC; Bav = nAV;
    }

    // Min+argmin across the 16 lanes of each half (masks 1..8 keep the wave32
    // halves separate); smallest-index tie-break = jnp.argmin first-match.
#pragma unroll
    for (int r = 0; r < 8; ++r) {
        float v = rmin[r]; int id = ridx[r];
#pragma unroll
        for (int m = 1; m < 16; m <<= 1) {
            float ov = __shfl_xor(v, m, 32);
            int oid = __shfl_xor(id, m, 32);
            bool take = (ov < v) | ((ov == v) & (oid < id));
            v = take ? ov : v; id = take ? oid : id;
        }
        if (lane == 0) {
            candD[seg * P + pbase + r] = v;
            candI[seg * P + pbase + r] = id;
        }
        if (lane == 16) {
            candD[seg * P + pbase + 8 + r] = v;
            candI[seg * P + pbase + 8 + r] = id;
        }
    }
}

// ---------------------------------------------------------------------------
// Combine the FSPLIT segment candidates per point (segments are ordered by
// triangle index, so lowest-segment-first tie-break preserves first-match).
// ---------------------------------------------------------------------------
__global__ void meshsdf_combine(const float* __restrict__ candD,
                                const int* __restrict__ candI,
                                float* __restrict__ dist_out,
                                int* __restrict__ assoc_out, int P) {
    int p = blockIdx.x * blockDim.x + threadIdx.x;
    if (p >= P) return;
    float v = candD[p]; int id = candI[p];
#pragma unroll
    for (int s = 1; s < FSPLIT; ++s) {
        float ov = candD[s * P + p];
        int oid = candI[s * P + p];
        bool take = (ov < v) | ((ov == v) & (oid < id));
        v = take ? ov : v; id = take ? oid : id;
    }
    dist_out[p] = v;
    assoc_out[p] = id;
}

// ---------------------------------------------------------------------------
// Loss: sum(dist) * (1000 / P) -> out[0]
// ---------------------------------------------------------------------------
__global__ void meshsdf_loss(float* __restrict__ out, int P) {
    __shared__ float sm[256];
    float s = 0.f;
    for (int i = threadIdx.x; i < P; i += 256) s += out[1 + i];
    sm[threadIdx.x] = s;
    __syncthreads();
    for (int k = 128; k > 0; k >>= 1) {
        if ((int)threadIdx.x < k) sm[threadIdx.x] += sm[threadIdx.x + k];
        __syncthreads();
    }
    if (threadIdx.x == 0) out[0] = sm[0] * (1000.0f / (float)P);
}

extern "C" void kernel_launch(void* const* d_in, const int* in_sizes, int n_in,
                              void* d_out, int out_size, void* d_ws, size_t ws_size,
                              hipStream_t stream) {
    const float* verts  = (const float*)d_in[0];
    const int*   faces  = (const int*)d_in[1];
    const float* points = (const float*)d_in[2];
    const int F = in_sizes[1] / 3;
    const int P = in_sizes[2] / 3;
    const int ntile = F >> 4;

    float* ws    = (float*)d_ws;
    float* wsAB  = ws;
    float* wsAC  = wsAB + (size_t)ntile * 64;
    float* wsAV  = wsAC + (size_t)ntile * 64;
    float* wsK   = wsAV + (size_t)ntile * 64;
    float* candD = wsK + (size_t)F * 8;
    int*   candI = (int*)(candD + (size_t)FSPLIT * P);

    float* out   = (float*)d_out;
    float* dist  = out + 1;
    int*   assoc = (int*)(out + 1 + P);

    meshsdf_prep<<<(F + 127) / 128, 128, 0, stream>>>(verts, faces, wsAB, wsAC,
                                                      wsAV, wsK, F);
    const int waves  = (P / 16) * FSPLIT;        // 16-point strip x F-segment
    const int blocks = (waves + 3) / 4;          // 4 waves (128 thr) per block
    meshsdf_main<<<blocks, 128, 0, stream>>>(points, wsAB, wsAC, wsAV, wsK,
                                             candD, candI, P, F);
    meshsdf_combine<<<(P + 255) / 256, 256, 0, stream>>>(candD, candI, dist,
                                                         assoc, P);
    meshsdf_loss<<<1, 256, 0, stream>>>(out, P);
}